// CapsClassifier_73237782331699
// MI455X (gfx1250) — compile-verified
//
#include <hip/hip_runtime.h>
#include <hip/hip_bf16.h>
#include <math.h>

// Problem constants (from reference)
#define B_TOT   256
#define N_CAPS  2048
#define C_OUT   10
#define D_IN    8
#define D_OUT   16
#define ITERS   3

// Tiling
#define BG          16                      // batch cols per workgroup (WMMA N)
#define NBG         (B_TOT / BG)            // 16 batch groups
#define ISLICES     16                      // i-slices (grid.y)
#define ISLICE_LEN  (N_CAPS / ISLICES)      // 128 capsules per WG
#define WAVES       8                       // 256 threads / 32
#define I_PER_WAVE  (ISLICE_LEN / WAVES)    // 16 capsules per wave
#define JO          (C_OUT * D_OUT)         // 160
#define SJ_ELEMS    (BG * JO)               // 2560 per WG
#define VSUM_ELEMS  (B_TOT * JO)            // 40960

typedef float v2f __attribute__((ext_vector_type(2)));
typedef float v8f __attribute__((ext_vector_type(8)));

__global__ __launch_bounds__(256) void caps_zero(float* __restrict__ p, int n) {
    int t = blockIdx.x * 256 + threadIdx.x;
    if (t < n) p[t] = 0.0f;
}

// Phase A (transposed WMMA): D[o, b] = W_i[16o x 8n] * u^T[8n x 16b].
// Each lane owns one batch element (b = lane%16) and 8 o-values
// (o = v + 8*half), so the routing dot needs only ONE xor-16 shuffle per j
// and the softmax runs once per thread entirely in registers.
__global__ __launch_bounds__(256) void caps_routeA(const float* __restrict__ u,
                                                   const float* __restrict__ W,
                                                   const float* __restrict__ vsum,
                                                   float* __restrict__ s_part) {
    const int bg   = blockIdx.x;        // 0..15 batch group
    const int isl  = blockIdx.y;        // 0..15 i-slice
    const int tid  = threadIdx.x;
    const int wave = tid >> 5;
    const int lane = tid & 31;
    const int lh   = lane >> 4;         // lane half (selects K pair / o half)
    const int l16  = lane & 15;         // A row (o) and B col (b) index

    __shared__ float s_lds[SJ_ELEMS];   // partial s accumulator [16b][10j][16o]
    for (int t = tid; t < SJ_ELEMS; t += 256) s_lds[t] = 0.0f;
    __syncthreads();

    // Preload vsum for this thread's batch element: vsreg[j][v] = vsum[b, j, v+8*lh]
    v8f vsreg[C_OUT];
    {
        const float* vp = vsum + (bg * BG + l16) * JO + 8 * lh;
#pragma unroll
        for (int j = 0; j < C_OUT; ++j)
#pragma unroll
            for (int v = 0; v < 8; ++v)
                vsreg[j][v] = vp[j * D_OUT + v];
    }

    v8f sacc[C_OUT];
#pragma unroll
    for (int j = 0; j < C_OUT; ++j)
        sacc[j] = (v8f){0.f, 0.f, 0.f, 0.f, 0.f, 0.f, 0.f, 0.f};

    const int i0    = isl * ISLICE_LEN + wave * I_PER_WAVE;
    const int ubase = (bg * BG + l16) * (N_CAPS * D_IN);

    for (int ii = 0; ii < I_PER_WAVE; ++ii) {
        const int i = i0 + ii;

        // ---- B operand: u^T[8n x 16b]; lanes 0-15: K={k0,k0+1}, 16-31: K={k0+2,k0+3}
        const float* up = u + ubase + i * D_IN;
        v2f ub0, ub1;
        ub0.x = up[2 * lh + 0]; ub0.y = up[2 * lh + 1];
        ub1.x = up[4 + 2 * lh + 0]; ub1.y = up[4 + 2 * lh + 1];

        v8f   uh[C_OUT];
        float bv[C_OUT];
        float mx = -3.0e38f;

#pragma unroll
        for (int j = 0; j < C_OUT; ++j) {
            // ---- A operand: W_i[16o x 8n], row M = lane%16 = o
            const float* wp = W + (((i * C_OUT + j) * D_OUT) + l16) * D_IN;
            v2f wa0, wa1;
            wa0.x = wp[2 * lh + 0]; wa0.y = wp[2 * lh + 1];
            wa1.x = wp[4 + 2 * lh + 0]; wa1.y = wp[4 + 2 * lh + 1];

            v8f acc = {0.f, 0.f, 0.f, 0.f, 0.f, 0.f, 0.f, 0.f};
            acc = __builtin_amdgcn_wmma_f32_16x16x4_f32(false, wa0, false, ub0,
                                                        (short)0, acc, false, false);
            acc = __builtin_amdgcn_wmma_f32_16x16x4_f32(false, wa1, false, ub1,
                                                        (short)0, acc, false, false);
            uh[j] = acc;   // D: M = o = v + 8*lh, N = b = lane%16

            // ---- agreement logit: 8 in-register FMAs + one xor-16 shuffle
            float p = 0.0f;
#pragma unroll
            for (int v = 0; v < 8; ++v)
                p = fmaf(uh[j][v], vsreg[j][v], p);
            p += __shfl_xor(p, 16, 32);
            bv[j] = p;
            mx = fmaxf(mx, p);
        }

        // ---- softmax over j (per-thread, register resident)
        float ssum = 0.0f;
#pragma unroll
        for (int j = 0; j < C_OUT; ++j) { bv[j] = __expf(bv[j] - mx); ssum += bv[j]; }
        const float inv = 1.0f / ssum;

        // ---- accumulate c_ij * u_hat into per-thread s
#pragma unroll
        for (int j = 0; j < C_OUT; ++j) {
            const float c = bv[j] * inv;
#pragma unroll
            for (int v = 0; v < 8; ++v)
                sacc[j][v] = fmaf(c, uh[j][v], sacc[j][v]);
        }
    }

    // ---- deterministic wave-ordered reduction into LDS
    for (int w = 0; w < WAVES; ++w) {
        if (wave == w) {
#pragma unroll
            for (int j = 0; j < C_OUT; ++j)
#pragma unroll
                for (int v = 0; v < 8; ++v)
                    s_lds[(l16 * C_OUT + j) * D_OUT + (v + 8 * lh)] += sacc[j][v];
        }
        __syncthreads();
    }

    // ---- write per-slice partials: s_part[isl][B][C][D_OUT]
    float* sp = s_part + isl * VSUM_ELEMS + bg * SJ_ELEMS;
    for (int t = tid; t < SJ_ELEMS; t += 256) sp[t] = s_lds[t];
}

// Phase B: sum i-slice partials (fixed order -> deterministic), squash,
// update vsum; final iteration writes v_j and logits to d_out.
__global__ __launch_bounds__(256) void caps_routeB(const float* __restrict__ s_part,
                                                   float* __restrict__ vsum,
                                                   float* __restrict__ out,
                                                   int last) {
    const int tid  = threadIdx.x;
    const int g    = tid >> 4;                  // 16 (b,j) groups per block
    const int o    = tid & 15;
    const int pair = blockIdx.x * 16 + g;       // 0..2559
    const int b    = pair / C_OUT;
    const int j    = pair % C_OUT;
    const int idx  = (b * C_OUT + j) * D_OUT + o;

    float sv = 0.0f;
#pragma unroll
    for (int isl = 0; isl < ISLICES; ++isl)
        sv += s_part[isl * VSUM_ELEMS + idx];

    float d = sv * sv;
    d += __shfl_xor(d, 1, 16);
    d += __shfl_xor(d, 2, 16);
    d += __shfl_xor(d, 4, 16);
    d += __shfl_xor(d, 8, 16);

    const float nu    = sqrtf(d);
    const float nrm   = fmaxf(nu, 1e-7f);
    const float scale = 1.0f - 1.0f / (1.0f + nrm * nrm);
    const float vv    = scale * (sv / nrm);

    if (last) {
        out[idx] = vv;                                      // v_j [B,C,16]
        if (o == 0)
            out[VSUM_ELEMS + b * C_OUT + j] = scale * (nu / nrm);   // logits [B,C]
    } else {
        vsum[idx] += vv;
    }
}

extern "C" void kernel_launch(void* const* d_in, const int* in_sizes, int n_in,
                              void* d_out, int out_size, void* d_ws, size_t ws_size,
                              hipStream_t stream) {
    const float* u = (const float*)d_in[0];   // [256, 2048, 8]
    const float* W = (const float*)d_in[1];   // [2048, 10, 16, 8]
    float* out = (float*)d_out;               // 40960 (v_j) + 2560 (logits)

    float* vsum   = (float*)d_ws;             // 40960 floats
    float* s_part = vsum + VSUM_ELEMS;        // 16 * 40960 floats

    caps_zero<<<(VSUM_ELEMS + 255) / 256, 256, 0, stream>>>(vsum, VSUM_ELEMS);

    dim3 gridA(NBG, ISLICES);
    for (int it = 0; it < ITERS; ++it) {
        caps_routeA<<<gridA, 256, 0, stream>>>(u, W, vsum, s_part);
        caps_routeB<<<(B_TOT * C_OUT) / 16, 256, 0, stream>>>(s_part, vsum, out,
                                                              (it == ITERS - 1) ? 1 : 0);
    }
}